// HLinear_85023172591876
// MI455X (gfx1250) — compile-verified
//
#include <hip/hip_runtime.h>
#include <hip/hip_bf16.h>

// Poincare fully-connected layer for MI455X (gfx1250, wave32).
// GEMM via V_WMMA_F32_16X16X4_F32; epilogue transcendentals via
// v_sqrt_f32 / v_log_f32 / v_exp_f32 / v_rcp_f32.

typedef __attribute__((ext_vector_type(2))) float v2f;
typedef __attribute__((ext_vector_type(8))) float v8f;

#define TIN 64
#define TOUT 64
#define BLOCK_THREADS 256
#define WAVES_PER_BLOCK 8
#define ROWS_PER_WAVE 16
#define ROWS_PER_BLOCK (WAVES_PER_BLOCK * ROWS_PER_WAVE)  // 128

__global__ __launch_bounds__(BLOCK_THREADS)
void poincare_fc_kernel(const float* __restrict__ x,
                        const float* __restrict__ z,
                        const float* __restrict__ bias,
                        float* __restrict__ out)
{
    // zT[n][k] = z[k][n]  (so B fragments are contiguous b64 in LDS)
    __shared__ float zT[TOUT * TIN];
    __shared__ float s_chzn[TOUT];  // cosh(2b)/z_norm
    __shared__ float s_sh[TOUT];    // sinh(2b)
    __shared__ float s_tzn[TOUT];   // 2*z_norm / ln(2)-free (used with log2/exp2)

    const int tid = threadIdx.x;

    // ---- stage z transposed into LDS (coalesced reads) ----
    for (int e = tid; e < TIN * TOUT; e += BLOCK_THREADS) {
        int k = e >> 6;        // e / TOUT
        int n = e & 63;        // e % TOUT
        zT[n * TIN + k] = z[e];
    }
    __syncthreads();

    // ---- per-OUT constants ----
    if (tid < TOUT) {
        float s = 0.f;
        #pragma unroll 8
        for (int k = 0; k < TIN; ++k) {
            float v = zT[tid * TIN + k];
            s = fmaf(v, v, s);
        }
        float zn = fmaxf(sqrtf(s), 1e-15f);
        float t = 2.0f * bias[tid];            // c = 1 -> 2*sqrt(c)*b = 2b
        s_chzn[tid] = coshf(t) / zn;
        s_sh[tid]   = sinhf(t);
        s_tzn[tid]  = 2.0f * zn;
    }
    __syncthreads();

    const int wave = tid >> 5;
    const int lane = tid & 31;
    const int half = lane >> 4;   // 0 or 1
    const int lmod = lane & 15;

    const long long rowbase = (long long)blockIdx.x * ROWS_PER_BLOCK
                            + (long long)wave * ROWS_PER_WAVE;
    const float* xrow = x + (rowbase + lmod) * TIN;

    v8f acc[4];
    acc[0] = v8f{}; acc[1] = v8f{}; acc[2] = v8f{}; acc[3] = v8f{};
    float rowsum = 0.f;

    // ---- GEMM: 16 K-steps of V_WMMA_F32_16X16X4_F32 over 4 N-tiles ----
    // A layout (16x4 f32): lane half 0 -> K = k+{0,1}, half 1 -> K = k+{2,3}, M = lane%16
    // B layout (4x16 f32): same K mapping, N = lane%16 (+ 16*ntile)
    #pragma unroll
    for (int k = 0; k < TIN; k += 4) {
        v2f a = *(const v2f*)(xrow + k + 2 * half);
        rowsum = fmaf(a.x, a.x, rowsum);
        rowsum = fmaf(a.y, a.y, rowsum);
        v2f b0 = *(const v2f*)(&zT[( 0 + lmod) * TIN + k + 2 * half]);
        v2f b1 = *(const v2f*)(&zT[(16 + lmod) * TIN + k + 2 * half]);
        v2f b2 = *(const v2f*)(&zT[(32 + lmod) * TIN + k + 2 * half]);
        v2f b3 = *(const v2f*)(&zT[(48 + lmod) * TIN + k + 2 * half]);
        acc[0] = __builtin_amdgcn_wmma_f32_16x16x4_f32(false, a, false, b0, (short)0, acc[0], false, false);
        acc[1] = __builtin_amdgcn_wmma_f32_16x16x4_f32(false, a, false, b1, (short)0, acc[1], false, false);
        acc[2] = __builtin_amdgcn_wmma_f32_16x16x4_f32(false, a, false, b2, (short)0, acc[2], false, false);
        acc[3] = __builtin_amdgcn_wmma_f32_16x16x4_f32(false, a, false, b3, (short)0, acc[3], false, false);
    }

    // combine the two half-wave partial ||x||^2 sums (lanes L and L^16 share a row)
    rowsum += __shfl_xor(rowsum, 16, 32);
    float lam = 2.0f / (1.0f - rowsum);   // conformal factor, c = 1

    // per-column constants for my 4 N-tiles
    float cA[4], cB[4], cT[4];
    #pragma unroll
    for (int t = 0; t < 4; ++t) {
        cA[t] = s_chzn[t * 16 + lmod];
        cB[t] = s_sh[t * 16 + lmod];
        cT[t] = s_tzn[t * 16 + lmod];
    }

    const float LOG2E_INV = 0.69314718055994530942f; // ln 2
    (void)LOG2E_INV;

    // ---- epilogue: C/D layout -> lane holds (M = i + 8*half, N = t*16 + lmod) ----
    #pragma unroll
    for (int i = 0; i < 8; ++i) {
        // broadcast lambda for row M = i + 8*half (lives in lane (M%16))
        float lm  = __shfl(lam, i + 8 * half, 32);
        float lm1 = lm - 1.0f;

        float ws[4];
        float s = 0.f;
        #pragma unroll
        for (int t = 0; t < 4; ++t) {
            float dot = acc[t][i];
            // g = (lam/zn)*dot*cosh(2b) - (lam-1)*sinh(2b)
            float g  = fmaf(lm * dot, cA[t], -lm1 * cB[t]);
            float ag = fabsf(g);
            // w = sinh(2*zn*asinh(g)) = sign(g) * 0.5*(u^{2zn} - u^{-2zn}),
            // u = |g| + sqrt(g^2+1)  (>= 1, so p >= 1)
            float u  = ag + __builtin_amdgcn_sqrtf(fmaf(ag, ag, 1.0f));
            float p  = __builtin_amdgcn_exp2f(cT[t] * __builtin_amdgcn_logf(u));
            float wv = copysignf(0.5f * (p - __builtin_amdgcn_rcpf(p)), g);
            ws[t] = wv;
            s = fmaf(wv, wv, s);
        }
        // reduce ||w||^2 across the 16 lanes of this half (xor stays within half)
        s += __shfl_xor(s, 1, 32);
        s += __shfl_xor(s, 2, 32);
        s += __shfl_xor(s, 4, 32);
        s += __shfl_xor(s, 8, 32);

        float scale = __builtin_amdgcn_rcpf(1.0f + __builtin_amdgcn_sqrtf(1.0f + s));
        long long rowoff = (rowbase + i + 8 * half) * TOUT;
        #pragma unroll
        for (int t = 0; t < 4; ++t)
            out[rowoff + t * 16 + lmod] = ws[t] * scale;
    }
}

extern "C" void kernel_launch(void* const* d_in, const int* in_sizes, int n_in,
                              void* d_out, int out_size, void* d_ws, size_t ws_size,
                              hipStream_t stream) {
    const float* x    = (const float*)d_in[0];
    const float* z    = (const float*)d_in[1];
    const float* bias = (const float*)d_in[2];
    float* out        = (float*)d_out;

    const int B = in_sizes[0] / TIN;            // 1048576
    const int blocks = B / ROWS_PER_BLOCK;      // 8192 (B is a multiple of 128)

    poincare_fc_kernel<<<blocks, BLOCK_THREADS, 0, stream>>>(x, z, bias, out);
}